// SNNForecaster_89060441850412
// MI455X (gfx1250) — compile-verified
//
#include <hip/hip_runtime.h>

typedef float v2f __attribute__((ext_vector_type(2)));
typedef float v4f __attribute__((ext_vector_type(4)));
typedef float v8f __attribute__((ext_vector_type(8)));
typedef unsigned int u32x4 __attribute__((ext_vector_type(4)));
typedef int i32x4 __attribute__((ext_vector_type(4)));
typedef int i32x8 __attribute__((ext_vector_type(8)));

#define B_    256
#define T_    128
#define C_    512
#define HID_  1024
#define HOR_  96
#define THR_  1.0f

#define HT        128                     // h-tile per workgroup
#define KC        64                      // K chunk staged in LDS
#define XS_STRIDE 68                      // padded LDS row stride (floats)
#define PS_STRIDE 132                     // padded proj-slab stride (floats)
#define TILE_FLOATS (128 * XS_STRIDE)     // 8704 floats = 34816 B per tile
#define BUF_FLOATS  (2 * TILE_FLOATS)     // X tile + W tile per buffer
#define LDS_BYTES   (2 * BUF_FLOATS * 4)  // double buffered: 139264 B

#if __has_builtin(__builtin_amdgcn_tensor_load_to_lds)
#define USE_TDM 1
#else
#define USE_TDM 0
#endif

#if USE_TDM
// One TDM 2D tile load: 128 rows x 64 cols of f32, global row stride 512 elems,
// into LDS at byte offset lds_off. Hardware pads 4 dwords after every 64 dwords
// -> LDS row stride 68 floats (bank-skewed), matching the ds_load fragment map.
__device__ __forceinline__ void tdm_tile(const float* gptr, unsigned lds_off) {
  unsigned long long ga = (unsigned long long)(size_t)gptr;
  u32x4 g0;
  g0[0] = 1u;                                     // count=1 (valid), user mode
  g0[1] = lds_off;                                // lds_addr (bytes)
  g0[2] = (unsigned)(ga & 0xffffffffu);           // global_addr[31:0]
  g0[3] = (unsigned)((ga >> 32) & 0x01ffffffu)    // global_addr[56:32]
        | (2u << 30);                             // type = 2 ("image")
  i32x8 g1;
  g1[0] = (2 << 16)      // data_size = 4B
        | (1 << 20)      // pad_enable
        | (5 << 22)      // pad_interval: 64 dwords
        | (3 << 25);     // pad_amount : 4 dwords
  g1[1] = (int)(((unsigned)C_ & 0xffffu) << 16);               // tensor_dim0 lo16 = 512
  g1[2] = (int)(((unsigned)C_ >> 16) |
                (((unsigned)T_ & 0xffffu) << 16));             // dim0 hi16 | tensor_dim1 lo16 = 128
  g1[3] = (int)(((unsigned)T_ >> 16) | ((unsigned)KC << 16));  // dim1 hi16 | tile_dim0 = 64
  g1[4] = (int)(unsigned)T_;                                   // tile_dim1 = 128, tile_dim2 = 0
  g1[5] = C_;                                                  // tensor_dim0_stride lo32 = 512
  g1[6] = 0;                                                   // stride hi16 | dim1_stride lo16
  g1[7] = 0;
  i32x4 gz4 = {0, 0, 0, 0};                  // groups 2/3 unused (2-D tensor)
  i32x8 gz8 = {0, 0, 0, 0, 0, 0, 0, 0};      // trailing group (clang-23 6-arg form)
  __builtin_amdgcn_tensor_load_to_lds(g0, g1, gz4, gz4, gz8, 0);
}
#endif

// -------- Kernel 1: fused proj GEMM (x @ W1^T + b1) + leaky scan over T ----
__global__ __launch_bounds__(256) void snn_proj_scan(
    const float* __restrict__ x,     // [B,T,C]
    const float* __restrict__ W1,    // [HID,C]
    const float* __restrict__ b1,    // [HID]
    const float* __restrict__ beta_p,// [1]
    float* __restrict__ memT)        // [B,HID]
{
  extern __shared__ float smem[];
  float* Ps = smem;                  // [T_][PS_STRIDE], reuses buffer 0 after GEMM

  const int blk = blockIdx.x;
  const int b   = blk >> 3;          // HID_/HT == 8 tiles per batch
  const int h0  = (blk & 7) * HT;

  const int tid  = threadIdx.x;
  const int w    = tid >> 5;         // wave 0..7 -> N columns [w*16, w*16+16)
  const int lane = tid & 31;
  const int ln   = lane & 15;
  const int kh   = lane >> 4;        // lane-half selects K pair

  v8f vz = {0.f,0.f,0.f,0.f,0.f,0.f,0.f,0.f};
  v8f acc[8];
  #pragma unroll
  for (int i = 0; i < 8; ++i) acc[i] = vz;

  auto stage = [&](int chunk, int bufsel) {
    const int kc0 = chunk * KC;
#if USE_TDM
    if (w == 0) {   // wave-uniform: one wave issues both DMA descriptors
      tdm_tile(x  + (size_t)b  * T_ * C_ + kc0,
               (unsigned)(bufsel * BUF_FLOATS * 4));
      tdm_tile(W1 + (size_t)h0 * C_ + kc0,
               (unsigned)(bufsel * BUF_FLOATS * 4 + TILE_FLOATS * 4));
    }
#else
    float* Xd = smem + bufsel * BUF_FLOATS;
    float* Wd = Xd + TILE_FLOATS;
    #pragma unroll
    for (int i = 0; i < 8; ++i) {
      int f = i * 256 + tid, row = f >> 4, c4 = (f & 15) << 2;
      *(v4f*)(Xd + row * XS_STRIDE + c4) =
          *(const v4f*)(x  + ((size_t)b * T_ + row) * C_ + kc0 + c4);
      *(v4f*)(Wd + row * XS_STRIDE + c4) =
          *(const v4f*)(W1 + ((size_t)(h0 + row)) * C_ + kc0 + c4);
    }
#endif
  };

  // Prologue: stage chunk 0 into buffer 0.
  stage(0, 0);
#if USE_TDM
  if (w == 0) __builtin_amdgcn_s_wait_tensorcnt(0);
#endif
  __syncthreads();

  for (int chunk = 0; chunk < C_ / KC; ++chunk) {
    const int cur = chunk & 1;
    if (chunk + 1 < C_ / KC) stage(chunk + 1, cur ^ 1);  // overlap DMA w/ compute

    const float* Xs = smem + cur * BUF_FLOATS;
    const float* Wf = Xs + TILE_FLOATS;
    for (int k = 0; k < KC; k += 4) {
      // B fragment (4x16, K x N): Bmat[k][n] = W1[h0+n][k]
      v2f bf = *(const v2f*)(Wf + (w * 16 + ln) * XS_STRIDE + k + 2 * kh);
      #pragma unroll
      for (int mt = 0; mt < 8; ++mt) {
        // A fragment (16x4, M x K): A[m][k] = x[b][mt*16+m][k]
        v2f af = *(const v2f*)(Xs + (mt * 16 + ln) * XS_STRIDE + k + 2 * kh);
        acc[mt] = __builtin_amdgcn_wmma_f32_16x16x4_f32(
            false, af, false, bf, (short)0, acc[mt], false, false);
      }
    }
#if USE_TDM
    if (w == 0) __builtin_amdgcn_s_wait_tensorcnt(0);    // chunk+1 DMA landed
#endif
    __syncthreads();
  }

  // Dump proj slab to LDS (reusing buffer space), then scan columns over T.
  #pragma unroll
  for (int mt = 0; mt < 8; ++mt) {
    #pragma unroll
    for (int r = 0; r < 8; ++r) {
      int m = mt * 16 + r + 8 * kh;                 // C/D layout: M = r + 8*laneHalf
      Ps[m * PS_STRIDE + w * 16 + ln] = acc[mt][r];
    }
  }
  __syncthreads();

  if (tid < HT) {
    const float bias = b1[h0 + tid];
    float beta = beta_p[0];
    beta = fminf(fmaxf(beta, 0.f), 1.f);
    float mem = 0.f;
    for (int t = 0; t < T_; ++t) {
      float inp   = Ps[t * PS_STRIDE + tid] + bias;
      float reset = (mem > THR_) ? THR_ : 0.f;      // reset based on PREVIOUS mem
      mem = beta * mem + inp - reset;
    }
    memT[(size_t)b * HID_ + h0 + tid] = mem;
  }
}

// -------- Kernel 2: out = mem_T @ Wh^T + bh  (256 x 1024 x 96) -------------
__global__ __launch_bounds__(256) void snn_head(
    const float* __restrict__ memT,  // [B,HID]
    const float* __restrict__ Wh,    // [HOR,HID]
    const float* __restrict__ bh,    // [HOR]
    float* __restrict__ out)         // [B,HOR]
{
  const int tid  = threadIdx.x;
  const int w    = tid >> 5;
  const int lane = tid & 31;
  const int ln   = lane & 15;
  const int kh   = lane >> 4;

  const int gw = blockIdx.x * 8 + w;   // 96 waves total, one 16x16 tile each
  const int mt = gw / 6;               // 16 M tiles
  const int nt = gw % 6;               // 6  N tiles
  const int m0 = mt * 16;
  const int n0 = nt * 16;

  v8f acc = {0.f,0.f,0.f,0.f,0.f,0.f,0.f,0.f};
  const float* aRow = memT + (size_t)(m0 + ln) * HID_;
  const float* bRow = Wh   + (size_t)(n0 + ln) * HID_;
  for (int k = 0; k < HID_; k += 4) {
    v2f af = *(const v2f*)(aRow + k + 2 * kh);
    v2f bf = *(const v2f*)(bRow + k + 2 * kh);
    acc = __builtin_amdgcn_wmma_f32_16x16x4_f32(
        false, af, false, bf, (short)0, acc, false, false);
  }
  const float bias = bh[n0 + ln];
  #pragma unroll
  for (int r = 0; r < 8; ++r) {
    int m = m0 + r + 8 * kh;
    out[(size_t)m * HOR_ + n0 + ln] = acc[r] + bias;
  }
}

extern "C" void kernel_launch(void* const* d_in, const int* in_sizes, int n_in,
                              void* d_out, int out_size, void* d_ws, size_t ws_size,
                              hipStream_t stream) {
  (void)in_sizes; (void)n_in; (void)out_size; (void)ws_size;
  const float* x    = (const float*)d_in[0];
  const float* W1   = (const float*)d_in[1];
  const float* b1   = (const float*)d_in[2];
  const float* Wh   = (const float*)d_in[3];
  const float* bh   = (const float*)d_in[4];
  const float* beta = (const float*)d_in[5];

  float* memT = (float*)d_ws;          // [B,HID] = 1 MB scratch
  float* out  = (float*)d_out;         // [B,HOR]

  snn_proj_scan<<<B_ * (HID_ / HT), 256, LDS_BYTES, stream>>>(x, W1, b1, beta, memT);
  snn_head<<<(96 / 8), 256, 0, stream>>>(memT, Wh, bh, out);
}